// MultiBoxLoss_55044300865700
// MI455X (gfx1250) — compile-verified
//
#include <hip/hip_runtime.h>
#include <hip/hip_bf16.h>
#include <math.h>

#define NCLS   21
#define BB     32
#define PP     32768
#define OO     50
#define THRESH 0.5f
#define NEGPOS 3

typedef float v2f __attribute__((ext_vector_type(2)));
typedef float v8f __attribute__((ext_vector_type(8)));
typedef unsigned long long u64;
typedef unsigned int u32;

__device__ __forceinline__ float wredf(float v) {
#pragma unroll
    for (int o = 16; o > 0; o >>= 1) v += __shfl_xor(v, o, 32);
    return v;
}
__device__ __forceinline__ int wredi(int v) {
#pragma unroll
    for (int o = 16; o > 0; o >>= 1) v += __shfl_xor(v, o, 32);
    return v;
}
__device__ __forceinline__ u64 shflx_u64(u64 v, int m) {
    u32 lo = (u32)v, hi = (u32)(v >> 32);
    lo = __shfl_xor(lo, m, 32);
    hi = __shfl_xor(hi, m, 32);
    return ((u64)hi << 32) | lo;
}

// ---------------------------------------------------------------- init
__global__ void k_init(u64* __restrict__ bp, int* __restrict__ num_pos,
                       float* __restrict__ topk, float* __restrict__ gacc) {
    int tid = blockIdx.x * blockDim.x + threadIdx.x;
    if (tid < BB * OO) bp[tid] = 0ull;
    if (tid < BB) { num_pos[tid] = 0; topk[tid] = 0.f; }
    if (tid < 2)  gacc[tid] = 0.f;
}

// ------------------------------------------- per-truth best prior (argmax over P)
__global__ void k_best_prior(const float4* __restrict__ priors,
                             const float*  __restrict__ targets,   // [B,O,5]
                             u64* __restrict__ bp) {               // [B,O] packed
    __shared__ float s_t[OO][4];
    const int b   = blockIdx.y;
    const int tid = threadIdx.x;
    if (tid < OO) {
        const float* tg = targets + ((size_t)b * OO + tid) * 5;
        s_t[tid][0] = tg[0]; s_t[tid][1] = tg[1];
        s_t[tid][2] = tg[2]; s_t[tid][3] = tg[3];
    }
    __syncthreads();
    const int p    = blockIdx.x * blockDim.x + tid;
    const int lane = tid & 31;
    float4 pr = priors[p];
    float px1 = pr.x - 0.5f * pr.z, py1 = pr.y - 0.5f * pr.w;
    float px2 = pr.x + 0.5f * pr.z, py2 = pr.y + 0.5f * pr.w;
    float area_b = pr.z * pr.w;
    for (int t = 0; t < OO; ++t) {
        float tx1 = s_t[t][0], ty1 = s_t[t][1], tx2 = s_t[t][2], ty2 = s_t[t][3];
        float iw = fmaxf(fminf(tx2, px2) - fmaxf(tx1, px1), 0.f);
        float ih = fmaxf(fminf(ty2, py2) - fmaxf(ty1, py1), 0.f);
        float inter  = iw * ih;
        float area_a = (tx2 - tx1) * (ty2 - ty1);
        float iou = inter / (area_a + area_b - inter);
        // pack: iou bits high, ~prior_idx low -> atomicMax picks max iou, ties -> lowest idx
        u64 pk = ((u64)__float_as_uint(iou) << 32) | (u32)(0xFFFFFFFFu - (u32)p);
#pragma unroll
        for (int o = 16; o > 0; o >>= 1) {
            u64 oth = shflx_u64(pk, o);
            pk = (oth > pk) ? oth : pk;
        }
        if (lane == 0) atomicMax(&bp[b * OO + t], pk);
    }
}

// ------------------------------------------------------------- main pass
// One wave handles 16 priors; 2 lanes per prior feed the 16x4 f32 WMMA A-layout.
__global__ void k_main(const float4* __restrict__ loc_data,   // [B,P,4]
                       const float*  __restrict__ conf_data,  // [B,P,21]
                       const float4* __restrict__ priors,     // [P,4]
                       const float*  __restrict__ targets,    // [B,O,5]
                       const u64*    __restrict__ bp,
                       float* __restrict__ ce_mine,           // [B,P]
                       int*   __restrict__ num_pos,           // [B]
                       float* __restrict__ gacc) {            // [0]=loss_l [1]=pos_ce
    __shared__ float s_t[OO][4];
    __shared__ float s_lab[OO];
    __shared__ int   s_bp[OO];
    __shared__ float s_sum[8][256];   // per-wave WMMA D staging

    const int b   = blockIdx.y;
    const int tid = threadIdx.x;
    if (tid < OO) {
        const float* tg = targets + ((size_t)b * OO + tid) * 5;
        s_t[tid][0] = tg[0]; s_t[tid][1] = tg[1];
        s_t[tid][2] = tg[2]; s_t[tid][3] = tg[3];
        s_lab[tid]  = tg[4];
        s_bp[tid]   = (int)(0xFFFFFFFFu - (u32)(bp[b * OO + tid] & 0xFFFFFFFFull));
    }
    __syncthreads();

    const int wid  = tid >> 5;
    const int lane = tid & 31;
    const int half = lane >> 4;       // 0: classes {4c,4c+1}, 1: {4c+2,4c+3}
    const int l16  = lane & 15;
    const int q    = (blockIdx.x * 8 + wid) * 16 + l16;   // prior index

    // ---- stream conf row, per-prior max via paired lanes ----
    const float* crow = conf_data + ((size_t)b * PP + q) * NCLS;
    __builtin_prefetch(crow + NCLS * 128, 0, 1);   // global_prefetch_b8 ahead of stream
    float v[12];
#pragma unroll
    for (int c = 0; c < 6; ++c) {
        int k0 = 4 * c + 2 * half;
        v[2*c]   = (k0     < NCLS) ? crow[k0]     : -INFINITY;
        v[2*c+1] = (k0 + 1 < NCLS) ? crow[k0 + 1] : -INFINITY;
    }
    float m = v[0];
#pragma unroll
    for (int i = 1; i < 12; ++i) m = fmaxf(m, v[i]);
    m = fmaxf(m, __shfl_xor(m, 16, 32));           // row max per prior

    // ---- class-sum of exp via V_WMMA_F32_16X16X4_F32 with ones-B ----
    v8f acc = {};
    v2f bones; bones[0] = 1.0f; bones[1] = 1.0f;
#pragma unroll
    for (int c = 0; c < 6; ++c) {
        v2f a;
        a[0] = __expf(v[2*c]     - m);             // A VGPR0: K0 (lanes 0-15) / K2 (16-31)
        a[1] = __expf(v[2*c + 1] - m);             // A VGPR1: K1 / K3
        acc = __builtin_amdgcn_wmma_f32_16x16x4_f32(false, a, false, bones,
                                                    (short)0, acc, false, false);
    }
#pragma unroll
    for (int r = 0; r < 8; ++r) s_sum[wid][r * 32 + lane] = acc[r];
    __syncthreads();
    // D layout: VGPR r, lanes 0-15 -> M=r ; lanes 16-31 -> M=8+r (row-sum replicated over N)
    float sumexp = s_sum[wid][(l16 & 7) * 32 + ((l16 & 8) ? 16 : 0)];
    float lse = m + __logf(sumexp);

    // ---- per-prior best truth (25 truths per lane half, shuffle-combined) ----
    float4 pr = priors[q];
    float px1 = pr.x - 0.5f * pr.z, py1 = pr.y - 0.5f * pr.w;
    float px2 = pr.x + 0.5f * pr.z, py2 = pr.y + 0.5f * pr.w;
    float area_b = pr.z * pr.w;
    float best = -1.f; int bidx = 0;
    for (int t = half * 25; t < half * 25 + 25; ++t) {
        float tx1 = s_t[t][0], ty1 = s_t[t][1], tx2 = s_t[t][2], ty2 = s_t[t][3];
        float iw = fmaxf(fminf(tx2, px2) - fmaxf(tx1, px1), 0.f);
        float ih = fmaxf(fminf(ty2, py2) - fmaxf(ty1, py1), 0.f);
        float inter  = iw * ih;
        float area_a = (tx2 - tx1) * (ty2 - ty1);
        float iou = inter / (area_a + area_b - inter);
        if (iou > best) { best = iou; bidx = t; }
    }
    if (half) bidx += 0;                // half1 covers t>=25 already
    {
        float ob = __shfl_xor(best, 16, 32);
        int   oi = __shfl_xor(bidx, 16, 32);
        if (ob > best || (ob == best && oi < bidx)) { best = ob; bidx = oi; }
    }
    // forced matches: best_prior override (ascending t -> last write wins)
    for (int t = 0; t < OO; ++t)
        if (s_bp[t] == q) { best = 2.0f; bidx = t; }

    // ---- losses (lanes 0-15 own the 16 priors) ----
    bool  act = (lane < 16);
    bool  pos = false;
    float ce = 0.f, sl1 = 0.f;
    if (act) {
        int conf = (best < THRESH) ? 0 : (int)(s_lab[bidx] + 1.0f);
        pos = conf > 0;
        float tl = crow[conf];                       // true logit
        ce = lse - tl;
        ce_mine[(size_t)b * PP + q] = pos ? 0.f : ce;
        if (pos) {
            float tx1 = s_t[bidx][0], ty1 = s_t[bidx][1];
            float tx2 = s_t[bidx][2], ty2 = s_t[bidx][3];
            float gcx = (0.5f * (tx1 + tx2) - pr.x) / (0.1f * pr.z);
            float gcy = (0.5f * (ty1 + ty2) - pr.y) / (0.1f * pr.w);
            float gw  = __logf((tx2 - tx1) / pr.z) * 5.0f;   // /0.2
            float gh  = __logf((ty2 - ty1) / pr.w) * 5.0f;
            float4 ld = loc_data[(size_t)b * PP + q];
            float d[4] = { ld.x - gcx, ld.y - gcy, ld.z - gw, ld.w - gh };
#pragma unroll
            for (int i = 0; i < 4; ++i) {
                float ad = fabsf(d[i]);
                sl1 += (ad < 1.f) ? 0.5f * d[i] * d[i] : ad - 0.5f;
            }
        }
    }
    int   np  = wredi(pos ? 1 : 0);
    float pce = wredf(pos ? ce : 0.f);
    float lsl = wredf(sl1);
    if (lane == 0) {
        if (np) atomicAdd(&num_pos[b], np);
        atomicAdd(&gacc[1], pce);
        atomicAdd(&gacc[0], lsl);
    }
}

// ---------------------------------- top-k sum per batch (4-pass radix select)
__global__ void k_topk(const float* __restrict__ ce_mine,
                       const int*   __restrict__ num_pos,
                       float* __restrict__ topk) {
    const int b   = blockIdx.x;
    const int tid = threadIdx.x;
    __shared__ int   hist[256];
    __shared__ u32   s_prefix;
    __shared__ int   s_rem;
    __shared__ float s_red[256];
    int k = num_pos[b] * NEGPOS;
    if (k > PP - 1) k = PP - 1;
    if (k <= 0) { if (tid == 0) topk[b] = 0.f; return; }
    const float* row = ce_mine + (size_t)b * PP;
    if (tid == 0) { s_prefix = 0u; s_rem = k; }
    __syncthreads();
    for (int shift = 24; shift >= 0; shift -= 8) {
        hist[tid] = 0;
        __syncthreads();
        u32 prefix = s_prefix;
        u32 himask = (shift == 24) ? 0u : (0xFFFFFFFFu << (shift + 8));
        for (int i = tid; i < PP; i += 256) {
            u32 u = __float_as_uint(row[i]);      // ce >= 0 -> bits order == value order
            if ((u & himask) == prefix) atomicAdd(&hist[(u >> shift) & 255], 1);
        }
        __syncthreads();
        if (tid == 0) {
            int rem = s_rem, cum = 0;
            for (int bin = 255; bin >= 0; --bin) {
                int c = hist[bin];
                if (cum + c >= rem) { s_prefix = prefix | ((u32)bin << shift); s_rem = rem - cum; break; }
                cum += c;
            }
        }
        __syncthreads();
    }
    u32 T   = s_prefix;                  // bits of k-th largest value
    int rem = s_rem;                     // how many tied-at-T elements included
    float sum = 0.f;
    for (int i = tid; i < PP; i += 256) {
        float x = row[i];
        if (__float_as_uint(x) > T) sum += x;
    }
    s_red[tid] = sum;
    __syncthreads();
    for (int s = 128; s > 0; s >>= 1) { if (tid < s) s_red[tid] += s_red[tid + s]; __syncthreads(); }
    if (tid == 0) topk[b] = s_red[0] + (float)rem * __uint_as_float(T);
}

// ----------------------------------------------------------------- finalize
__global__ void k_final(const int* __restrict__ num_pos, const float* __restrict__ topk,
                        const float* __restrict__ gacc, float* __restrict__ out) {
    if (threadIdx.x == 0) {
        int N = 0; float lc = gacc[1];
        for (int b = 0; b < BB; ++b) { N += num_pos[b]; lc += topk[b]; }
        float fN = (float)N;
        out[0] = gacc[0] / fN;
        out[1] = lc / fN;
    }
}

extern "C" void kernel_launch(void* const* d_in, const int* in_sizes, int n_in,
                              void* d_out, int out_size, void* d_ws, size_t ws_size,
                              hipStream_t stream) {
    const float* loc     = (const float*)d_in[0];   // [B,P,4]
    const float* conf    = (const float*)d_in[1];   // [B,P,21]
    const float* priors  = (const float*)d_in[2];   // [P,4]
    const float* targets = (const float*)d_in[3];   // [B,O,5]

    char* ws = (char*)d_ws;
    float* ce_mine = (float*)ws;                                        // B*P floats
    u64*   bpv     = (u64*)(ws + (size_t)BB * PP * sizeof(float));      // B*O u64 (8-aligned)
    int*   np      = (int*)((char*)bpv + (size_t)BB * OO * sizeof(u64));
    float* topk    = (float*)((char*)np + BB * sizeof(int));
    float* gacc    = topk + BB;

    k_init<<<(BB * OO + 255) / 256, 256, 0, stream>>>(bpv, np, topk, gacc);
    k_best_prior<<<dim3(PP / 256, BB), 256, 0, stream>>>((const float4*)priors, targets, bpv);
    k_main<<<dim3(PP / 128, BB), 256, 0, stream>>>((const float4*)loc, conf,
                                                   (const float4*)priors, targets,
                                                   bpv, ce_mine, np, gacc);
    k_topk<<<BB, 256, 0, stream>>>(ce_mine, np, topk);
    k_final<<<1, 32, 0, stream>>>(np, topk, gacc, (float*)d_out);
}